// SAGEConv_61220463837398
// MI455X (gfx1250) — compile-verified
//
#include <hip/hip_runtime.h>

typedef __attribute__((ext_vector_type(2))) float v2f;
typedef __attribute__((ext_vector_type(8))) float v8f;

#define N_NODES 50000
#define N_EDGES 600000
#define F       128
#define LPAD    132   // padded LDS row stride (floats): multiple of 4 -> every
                      // 16B chunk is 16B-aligned for b128 async stores;
                      // 132 mod 64 = 4 -> 16 consecutive rows hit distinct banks

// ---- CDNA5 async global->LDS copy (ASYNCcnt-tracked, no VGPR round trip) ----
__device__ __forceinline__ unsigned lds_off(const void* p) {
    // generic pointer to __shared__: low 32 bits are the LDS byte address
    return (unsigned)(unsigned long long)p;
}
__device__ __forceinline__ void async_copy16(unsigned lds_byte_off, const float* g) {
    asm volatile("global_load_async_to_lds_b128 %0, %1, off"
                 :: "v"(lds_byte_off), "v"(g)
                 : "memory");
}
__device__ __forceinline__ void wait_async0() {
    asm volatile("s_wait_asynccnt 0x0" ::: "memory");
}

// ---------------------------------------------------------------------------
// Kernel 1: zero the aggregation buffer (d_out doubles as agg) and degrees.
// ---------------------------------------------------------------------------
__global__ void sage_init_kernel(float* __restrict__ agg, float* __restrict__ deg) {
    long long i = (long long)blockIdx.x * blockDim.x + threadIdx.x;
    if (i < (long long)N_NODES * F) agg[i] = 0.0f;
    if (i < N_NODES)                deg[i] = 0.0f;
}

// ---------------------------------------------------------------------------
// Kernel 2: edge scatter. 8 edges per 256-thread block; 32 lanes x float4 per
// edge. x (25.6 MB) is L2-resident (192 MB L2) so gathers are L2 hits; the
// scatter is 76.8M f32 atomics resolved in L2's atomic units.
// ---------------------------------------------------------------------------
__global__ void sage_scatter_kernel(const float* __restrict__ x,
                                    const int*   __restrict__ src,
                                    const int*   __restrict__ dst,
                                    float*       __restrict__ agg,
                                    float*       __restrict__ deg) {
    const int e    = blockIdx.x * 8 + (threadIdx.x >> 5);   // exact: 75000*8
    const int c    = (threadIdx.x & 31) << 2;               // feature base
    const int s    = src[e];
    const int d    = dst[e];
    const float4 v = *(const float4*)&x[(long long)s * F + c];
    float* a = &agg[(long long)d * F + c];
    atomicAdd(a + 0, v.x);
    atomicAdd(a + 1, v.y);
    atomicAdd(a + 2, v.z);
    atomicAdd(a + 3, v.w);
    if ((threadIdx.x & 31) == 0) atomicAdd(&deg[d], 1.0f);
}

// ---------------------------------------------------------------------------
// Kernel 3: fused  out = relu(x @ Wself + (agg/max(deg,1)) @ Wneigh + b)
// V_WMMA_F32_16X16X4_F32, 8 waves/block (one output column tile each),
// weights transposed in LDS, node tiles double-buffered via async-to-LDS.
// ---------------------------------------------------------------------------
__global__ void __launch_bounds__(256)
sage_gemm_kernel(const float* __restrict__ x,
                 const float* __restrict__ Wself,
                 const float* __restrict__ Wneigh,
                 const float* __restrict__ bias,
                 const float* __restrict__ deg,
                 float*       __restrict__ out /* agg in, result out */) {
    __shared__ float WsT[F * LPAD];        // WsT[n*LPAD + k] = Wself[k*F + n]
    __shared__ float WnT[F * LPAD];
    __shared__ float xs[2][16 * LPAD];     // double-buffered x tile
    __shared__ float hs[2][16 * LPAD];     // double-buffered raw-agg tile

    const int tid    = threadIdx.x;
    const int nTiles = N_NODES / 16;       // 3125, exact

    // Prologue: kick off async staging of this block's first tile.
    int tile = blockIdx.x;
    int buf  = 0;
    if (tile < nTiles) {
        const long long base = (long long)tile * 16 * F;
        #pragma unroll
        for (int c = tid; c < 512; c += 256) {          // 512 x 16B chunks
            int m = c >> 5;
            int k = (c & 31) << 2;
            async_copy16(lds_off(&xs[0][m * LPAD + k]), &x  [base + c * 4]);
            async_copy16(lds_off(&hs[0][m * LPAD + k]), &out[base + c * 4]);
        }
    }

    // Load + transpose both weight matrices into LDS (one-time, coalesced).
    for (int i = tid; i < F * F; i += 256) {
        int k = i >> 7;          // row of W (input feature)
        int n = i & (F - 1);     // col of W (output feature)
        WsT[n * LPAD + k] = Wself[i];
        WnT[n * LPAD + k] = Wneigh[i];
    }

    const int lane = tid & 31;            // wave32
    const int wv   = tid >> 5;            // 0..7 -> output column tile
    const int mrow = lane & 15;           // A row / B col / C-D col
    const int koff = (lane >> 4) << 1;    // 0 or 2: K sub-offset per half-wave
    const float bval = bias[wv * 16 + mrow];

    for (; tile < nTiles; tile += gridDim.x) {
        const int r0 = tile * 16;

        // Current tile's async loads done; also fences weight staging (1st
        // iter) and guarantees everyone finished reading the other buffer.
        wait_async0();
        __syncthreads();

        // Prefetch next tile into the other buffer while we compute.
        const int nxt = tile + gridDim.x;
        if (nxt < nTiles) {
            const long long base = (long long)nxt * 16 * F;
            #pragma unroll
            for (int c = tid; c < 512; c += 256) {
                int m = c >> 5;
                int k = (c & 31) << 2;
                async_copy16(lds_off(&xs[buf ^ 1][m * LPAD + k]), &x  [base + c * 4]);
                async_copy16(lds_off(&hs[buf ^ 1][m * LPAD + k]), &out[base + c * 4]);
            }
        }

        // Per-row mean scale, folded into the neighbor A operand.
        const float sc = 1.0f / fmaxf(deg[r0 + mrow], 1.0f);

        const float* xrow = &xs[buf][mrow * LPAD + koff];
        const float* hrow = &hs[buf][mrow * LPAD + koff];
        const float* wsr  = &WsT[mrow * LPAD + koff];
        const float* wnr  = &WnT[mrow * LPAD + koff];

        v8f cs = {bval, bval, bval, bval, bval, bval, bval, bval};
        v8f cn = {0.f, 0.f, 0.f, 0.f, 0.f, 0.f, 0.f, 0.f};
        #pragma unroll
        for (int k = 0; k < F; k += 4) {
            v2f ax = *(const v2f*)&xrow[k];
            v2f ah = *(const v2f*)&hrow[k];
            ah *= sc;
            v2f bs = *(const v2f*)&wsr[k];
            v2f bn = *(const v2f*)&wnr[k];
            cs = __builtin_amdgcn_wmma_f32_16x16x4_f32(
                     false, ax, false, bs, (short)0, cs, false, false);
            cn = __builtin_amdgcn_wmma_f32_16x16x4_f32(
                     false, ah, false, bn, (short)0, cn, false, false);
        }

        // D layout: VGPR r -> row r0 + r + (lane>=16 ? 8 : 0), col = wv*16+mrow
        const int colBase = wv * 16 + mrow;
        const int rAdd    = (lane >> 4) << 3;
        #pragma unroll
        for (int r = 0; r < 8; r++) {
            int row = r0 + r + rAdd;
            out[(long long)row * F + colBase] = fmaxf(cs[r] + cn[r], 0.0f);
        }
        buf ^= 1;
    }
}

// ---------------------------------------------------------------------------
extern "C" void kernel_launch(void* const* d_in, const int* in_sizes, int n_in,
                              void* d_out, int out_size, void* d_ws, size_t ws_size,
                              hipStream_t stream) {
    const float* x      = (const float*)d_in[0];
    const float* Wself  = (const float*)d_in[1];
    const float* Wneigh = (const float*)d_in[2];
    const float* bias   = (const float*)d_in[3];
    const int*   src    = (const int*)d_in[4];
    const int*   dst    = (const int*)d_in[5];
    float* out = (float*)d_out;
    float* deg = (float*)d_ws;   // N_NODES floats = 200 KB of workspace

    {
        long long total = (long long)N_NODES * F;
        int blocks = (int)((total + 255) / 256);
        sage_init_kernel<<<blocks, 256, 0, stream>>>(out, deg);
    }
    {
        sage_scatter_kernel<<<N_EDGES / 8, 256, 0, stream>>>(x, src, dst, out, deg);
    }
    {
        sage_gemm_kernel<<<512, 256, 0, stream>>>(x, Wself, Wneigh, bias, deg, out);
    }
}